// GNNScalablePredictor_69887707840666
// MI455X (gfx1250) — compile-verified
//
#include <hip/hip_runtime.h>

typedef __attribute__((ext_vector_type(2))) float v2f;
typedef __attribute__((ext_vector_type(8))) float v8f;

// ---------------------------------------------------------------------------
// Degree accumulation: deg[dst[e]] += 1  (hardware f32 atomic, L2-resident)
// ---------------------------------------------------------------------------
__global__ __launch_bounds__(256) void deg_accum_kernel(
    const long long* __restrict__ dst, float* __restrict__ deg, int nE) {
  int e = blockIdx.x * blockDim.x + threadIdx.x;
  if (e >= nE) return;
  unsafeAtomicAdd(&deg[(size_t)dst[e]], 1.0f);
}

__global__ __launch_bounds__(256) void inv_deg_kernel(float* __restrict__ deg, int nN) {
  int i = blockIdx.x * blockDim.x + threadIdx.x;
  if (i >= nN) return;
  deg[i] = 1.0f / fmaxf(deg[i], 1.0f);
}

// ---------------------------------------------------------------------------
// Y = X @ W.T via V_WMMA_F32_16X16X4_F32 (full fp32 matrix pipe).
//   X: [n_rows, K] row-major, W: [64, K] row-major, Y: [n_rows, 64].
// Block = 128 threads = 4 waves; each wave owns a 16x16 tile of the 16x64
// output strip for this block. Two accumulator chains hide WMMA RAW latency.
// A frag (16x4 f32): lane l -> row = l%16; VGPR0 = K+2*(l/16), VGPR1 = +1.
// B frag (4x16 f32): lane l -> col = l%16; same K split across lane halves.
// C/D (16x16 f32):   VGPR v, lane l -> row = v + 8*(l/16), col = l%16.
// ---------------------------------------------------------------------------
template <int K>
__global__ __launch_bounds__(128) void gemm_xwT_kernel(
    const float* __restrict__ X, const float* __restrict__ W,
    float* __restrict__ Y, int n_rows) {
  const int lane  = threadIdx.x & 31;
  const int wave  = threadIdx.x >> 5;       // N-tile: columns [16*wave, 16*wave+16)
  const int m     = lane & 15;
  const int khalf = lane >> 4;              // 0 or 1 -> K offset 0 / 2
  const int row0  = blockIdx.x * 16;

  int arow = row0 + m;
  if (arow >= n_rows) arow = n_rows - 1;    // clamp (N is a multiple of 16 anyway)
  const int bcol = wave * 16 + m;           // row of W == column of W.T

  const float* __restrict__ Xrow = X + (size_t)arow * K + 2 * khalf;
  const float* __restrict__ Wrow = W + (size_t)bcol * K + 2 * khalf;

  v8f c0 = {};
  v8f c1 = {};
#pragma unroll
  for (int k = 0; k < K; k += 8) {
    v2f a0 = *(const v2f*)(Xrow + k);
    v2f b0 = *(const v2f*)(Wrow + k);
    c0 = __builtin_amdgcn_wmma_f32_16x16x4_f32(false, a0, false, b0,
                                               (short)0, c0, false, false);
    v2f a1 = *(const v2f*)(Xrow + k + 4);
    v2f b1 = *(const v2f*)(Wrow + k + 4);
    c1 = __builtin_amdgcn_wmma_f32_16x16x4_f32(false, a1, false, b1,
                                               (short)0, c1, false, false);
  }
#pragma unroll
  for (int v = 0; v < 8; ++v) c0[v] += c1[v];

  const int crow0 = row0 + 8 * khalf;
  const int ccol  = wave * 16 + m;
  float* __restrict__ Yp = Y + (size_t)crow0 * 64 + ccol;
#pragma unroll
  for (int v = 0; v < 8; ++v) {
    if (crow0 + v < n_rows) Yp[(size_t)v * 64] = c0[v];
  }
}

// ---------------------------------------------------------------------------
// Edge scatter: agg[dst[e], :] += Y[src[e], :]  (64 f32 per edge).
// 16 threads per edge, float4 each -> coalesced 16B gathers (L2-resident),
// hardware global_atomic_add_f32 accumulation.
// ---------------------------------------------------------------------------
__global__ __launch_bounds__(256) void edge_scatter_kernel(
    const long long* __restrict__ src, const long long* __restrict__ dst,
    const float* __restrict__ Y, float* __restrict__ agg, int nE) {
  long long tid = (long long)blockIdx.x * blockDim.x + threadIdx.x;
  long long e = tid >> 4;
  if (e >= nE) return;
  const int f = (int)(tid & 15) << 2;
  const long long s = src[e];
  const long long d = dst[e];
  const float4 v = *(const float4*)(Y + (size_t)s * 64 + f);
  float* __restrict__ ap = agg + (size_t)d * 64 + f;
  unsafeAtomicAdd(ap + 0, v.x);
  unsafeAtomicAdd(ap + 1, v.y);
  unsafeAtomicAdd(ap + 2, v.z);
  unsafeAtomicAdd(ap + 3, v.w);
}

// ---------------------------------------------------------------------------
// h = relu(agg * inv_deg + bias + y_r)   elementwise, float4-vectorized
// ---------------------------------------------------------------------------
__global__ __launch_bounds__(256) void combine_relu_kernel(
    const float* __restrict__ agg, const float* __restrict__ yr,
    const float* __restrict__ inv_deg, const float* __restrict__ bias,
    float* __restrict__ h, int nN) {
  long long tid = (long long)blockIdx.x * blockDim.x + threadIdx.x;
  long long n = tid >> 4;
  if (n >= nN) return;
  const int f = (int)(tid & 15) << 2;
  const float id = inv_deg[n];
  const float4 a = *(const float4*)(agg + (size_t)n * 64 + f);
  const float4 r = *(const float4*)(yr + (size_t)n * 64 + f);
  const float4 b = *(const float4*)(bias + f);
  float4 o;
  o.x = fmaxf(fmaf(a.x, id, b.x) + r.x, 0.0f);
  o.y = fmaxf(fmaf(a.y, id, b.y) + r.y, 0.0f);
  o.z = fmaxf(fmaf(a.z, id, b.z) + r.z, 0.0f);
  o.w = fmaxf(fmaf(a.w, id, b.w) + r.w, 0.0f);
  *(float4*)(h + (size_t)n * 64 + f) = o;
}

// ---------------------------------------------------------------------------
// Fused layer-2 combine + MLP head:
//   h2 = relu(agg * inv_deg + b2_l + y2r)            [64]
//   p  = relu(h2 @ Wp1.T + bp1)                      [32]
//   out = p @ Wp2.T + bp2                            [1]
// One thread per node; MLP weights staged in LDS (broadcast reads).
// ---------------------------------------------------------------------------
__global__ __launch_bounds__(256) void final_mlp_kernel(
    const float* __restrict__ agg, const float* __restrict__ yr,
    const float* __restrict__ inv_deg, const float* __restrict__ b2l,
    const float* __restrict__ Wp1, const float* __restrict__ bp1,
    const float* __restrict__ Wp2, const float* __restrict__ bp2,
    float* __restrict__ out, int nN) {
  __shared__ float sW1[32 * 64];
  __shared__ float sb1[32];
  __shared__ float sW2[32];
  __shared__ float sB2[64];
  __shared__ float sb2;
  for (int i = threadIdx.x; i < 32 * 64; i += 256) sW1[i] = Wp1[i];
  if (threadIdx.x < 32) {
    sb1[threadIdx.x] = bp1[threadIdx.x];
    sW2[threadIdx.x] = Wp2[threadIdx.x];
  }
  if (threadIdx.x < 64) sB2[threadIdx.x] = b2l[threadIdx.x];
  if (threadIdx.x == 0) sb2 = bp2[0];
  __syncthreads();

  const int n = blockIdx.x * 256 + threadIdx.x;
  if (n >= nN) return;

  const float id = inv_deg[n];
  const float* __restrict__ ap = agg + (size_t)n * 64;
  const float* __restrict__ rp = yr + (size_t)n * 64;
  float h2[64];
#pragma unroll
  for (int j = 0; j < 64; ++j)
    h2[j] = fmaxf(fmaf(ap[j], id, sB2[j]) + rp[j], 0.0f);

  float acc = sb2;
  for (int j = 0; j < 32; ++j) {
    float p = sb1[j];
    const float* __restrict__ wrow = &sW1[j * 64];
#pragma unroll
    for (int k = 0; k < 64; ++k) p = fmaf(h2[k], wrow[k], p);
    acc = fmaf(fmaxf(p, 0.0f), sW2[j], acc);
  }
  out[n] = acc;
}

// ---------------------------------------------------------------------------
// Host side
// ---------------------------------------------------------------------------
extern "C" void kernel_launch(void* const* d_in, const int* in_sizes, int n_in,
                              void* d_out, int out_size, void* d_ws, size_t ws_size,
                              hipStream_t stream) {
  const float*     x    = (const float*)d_in[0];
  const long long* ei   = (const long long*)d_in[1];
  const float*     W1_l = (const float*)d_in[2];
  const float*     b1_l = (const float*)d_in[3];
  const float*     W1_r = (const float*)d_in[4];
  const float*     W2_l = (const float*)d_in[5];
  const float*     b2_l = (const float*)d_in[6];
  const float*     W2_r = (const float*)d_in[7];
  const float*     Wp1  = (const float*)d_in[8];
  const float*     bp1  = (const float*)d_in[9];
  const float*     Wp2  = (const float*)d_in[10];
  const float*     bp2  = (const float*)d_in[11];
  float* out = (float*)d_out;

  const int nN = in_sizes[0] / 128;      // 100000
  const int nE = in_sizes[1] / 2;        // 3200000
  const long long* src = ei;             // edge_index[0, :]
  const long long* dst = ei + nE;        // edge_index[1, :]

  const size_t N64 = (size_t)nN * 64;
  float* inv_deg = (float*)d_ws;         // [N]
  float* buf_l   = inv_deg + nN;         // [N,64]  y1l, later y2l
  float* buf_r   = buf_l + N64;          // [N,64]  y1r, later y2r
  float* buf_agg = buf_r + N64;          // [N,64]  aggregation target
  float* buf_h   = buf_agg + N64;        // [N,64]  h1

  const int mtiles   = (nN + 15) / 16;
  const int eblocks  = (int)(((long long)nE * 16 + 255) / 256);
  const int nblocks4 = (int)(((long long)nN * 16 + 255) / 256);

  // Degree / inverse degree
  hipMemsetAsync(inv_deg, 0, sizeof(float) * nN, stream);
  deg_accum_kernel<<<(nE + 255) / 256, 256, 0, stream>>>(dst, inv_deg, nE);
  inv_deg_kernel<<<(nN + 255) / 256, 256, 0, stream>>>(inv_deg, nN);

  // Layer 1: transform first (128 -> 64) so the edge pass moves half the data
  gemm_xwT_kernel<128><<<mtiles, 128, 0, stream>>>(x, W1_l, buf_l, nN);
  gemm_xwT_kernel<128><<<mtiles, 128, 0, stream>>>(x, W1_r, buf_r, nN);
  hipMemsetAsync(buf_agg, 0, sizeof(float) * N64, stream);
  edge_scatter_kernel<<<eblocks, 256, 0, stream>>>(src, dst, buf_l, buf_agg, nE);
  combine_relu_kernel<<<nblocks4, 256, 0, stream>>>(buf_agg, buf_r, inv_deg, b1_l,
                                                    buf_h, nN);

  // Layer 2
  gemm_xwT_kernel<64><<<mtiles, 128, 0, stream>>>(buf_h, W2_l, buf_l, nN);
  gemm_xwT_kernel<64><<<mtiles, 128, 0, stream>>>(buf_h, W2_r, buf_r, nN);
  hipMemsetAsync(buf_agg, 0, sizeof(float) * N64, stream);
  edge_scatter_kernel<<<eblocks, 256, 0, stream>>>(src, dst, buf_l, buf_agg, nE);

  // Fused combine + MLP head
  final_mlp_kernel<<<(nN + 255) / 256, 256, 0, stream>>>(
      buf_agg, buf_r, inv_deg, b2_l, Wp1, bp1, Wp2, bp2, out, nN);
}